// LiftSplatShoot_14611478741113
// MI455X (gfx1250) — compile-verified
//
#include <hip/hip_runtime.h>
#include <cstdint>

// ---------------- problem constants (from reference config) ----------------
#define NPTS   692736          // B*N*D*H*W = 4*6*41*16*44
#define PPB    173184          // points per batch (NPTS / 4)
#define NC     64              // channels
#define NXY    200             // grid X = grid Y
#define VPB    40000           // voxels per batch (200*200), Z collapsed (nx2==1)
#define NVOX   160000          // 4 * 40000
#define CAP    32              // fixed bin capacity (mean occupancy ~3.7)
#define OVF    0x80000000u
#define DROPPED 0x7FFFFFFFu    // OVF bit clear

typedef unsigned int v4u __attribute__((ext_vector_type(4)));
typedef int          v8i __attribute__((ext_vector_type(8)));
typedef int          v4i __attribute__((ext_vector_type(4)));

#if defined(__has_builtin)
#if __has_builtin(__builtin_amdgcn_tensor_load_to_lds)
#define HAVE_TDM 1
#endif
#endif

// ---------------- kernel: per-voxel segment reduce (one wave32 / voxel) ----------------
// FIRST in the file so the disasm snippet shows the TDM gather path.
// Gathers each voxel's feature rows via the CDNA5 Tensor Data Mover in gather
// mode (up to 8 x 32-bit row indices per descriptor, 256B per row -> LDS),
// sums them, and writes the channel-major output with plain stores (no atomics;
// also serves as the zero-init of d_out for empty voxels).
__global__ __launch_bounds__(256) void lss_gather_reduce(const float* __restrict__ x,
                                                         const uint32_t* __restrict__ counts,
                                                         const uint32_t* __restrict__ bins,
                                                         float* __restrict__ out) {
    __shared__ float sbuf[8 * 8 * NC];        // 8 waves x 8 staged rows x 64 ch = 16 KB
    const int wave = threadIdx.x >> 5;
    const int lane = threadIdx.x & 31;        // wave32
    const int v = blockIdx.x * 8 + wave;
    if (v >= NVOX) return;

    uint32_t k = counts[v];
    if (k > CAP) k = CAP;
    const uint32_t* __restrict__ bin = bins + (size_t)v * CAP;

    float a0 = 0.0f, a1 = 0.0f;               // this lane's two channels: 2*lane, 2*lane+1

#if defined(HAVE_TDM)
    float* sw = sbuf + wave * (8 * NC);
    const uint32_t lds_off = (uint32_t)(uintptr_t)sw;         // low 32 bits = LDS byte offset
    const uint64_t gaddr   = (uint64_t)(uintptr_t)x;

    // ---- loop-invariant D# words (CDNA5 ISA) ----
    v4u g0;
    g0[0] = 0xC0000001u;                                      // count=1, 32-bit idx, gather_mode
    g0[1] = lds_off;                                          // lds_addr
    g0[2] = (uint32_t)gaddr;                                  // global_addr[31:0]
    g0[3] = ((uint32_t)(gaddr >> 32) & 0x01FFFFFFu) | 0x80000000u; // addr[56:32] | type=2
    v8i g1;
    g1[0] = (int)(2u << 16);                                  // data_size = 4 bytes
    g1[1] = (int)((uint32_t)NC << 16);                        // tensor_dim0 = 64
    g1[2] = (int)(((uint32_t)NPTS & 0xFFFFu) << 16);          // tensor_dim1 lo16
    g1[3] = (int)(((uint32_t)NPTS >> 16) | ((uint32_t)NC << 16)); // dim1 hi16 | tile_dim0=64
    g1[5] = NC;                                               // tensor_dim0_stride = 64
    g1[6] = 0;
    g1[7] = 0;

    for (uint32_t base = 0; base < k; base += 8u) {
        const uint32_t kk = (k - base < 8u) ? (k - base) : 8u;
        int idx[8];
#pragma unroll
        for (int j = 0; j < 8; ++j) {
            uint32_t id = (j < (int)kk) ? bin[base + (uint32_t)j] : 0u;
            idx[j] = __builtin_amdgcn_readfirstlane((int)id); // wave-uniform for D#
        }
        g1[4] = (int)kk;                                      // tile_dim1 = #valid indices
        v4i g2; g2[0] = idx[0]; g2[1] = idx[1]; g2[2] = idx[2]; g2[3] = idx[3];
        v4i g3; g3[0] = idx[4]; g3[1] = idx[5]; g3[2] = idx[6]; g3[3] = idx[7];
#if __clang_major__ >= 23
        v8i gz; gz[0]=0; gz[1]=0; gz[2]=0; gz[3]=0; gz[4]=0; gz[5]=0; gz[6]=0; gz[7]=0;
        __builtin_amdgcn_tensor_load_to_lds(g0, g1, g2, g3, gz, 0);
#else
        __builtin_amdgcn_tensor_load_to_lds(g0, g1, g2, g3, 0);
#endif
        __builtin_amdgcn_s_wait_tensorcnt(0);
        __asm__ volatile("" ::: "memory");                    // LDS now holds kk rows

        const float2* s2 = (const float2*)sw;
        for (uint32_t r = 0; r < kk; ++r) {
            float2 f = s2[r * 32 + (uint32_t)lane];           // conflict-free b64 LDS reads
            a0 += f.x; a1 += f.y;
        }
    }
#else
    // Fallback: direct coalesced row loads (256B/row, float2 per lane)
    for (uint32_t r = 0; r < k; ++r) {
        const float2* row = (const float2*)(x + (size_t)bin[r] * NC);
        float2 f = row[lane];
        a0 += f.x; a1 += f.y;
    }
#endif

    // out[b, c, gx, gy]  with flat voxel v = b*40000 + gx*200 + gy
    const uint32_t b  = (uint32_t)v / VPB;
    const uint32_t sp = (uint32_t)v % VPB;
    float* o = out + (size_t)b * ((size_t)NC * VPB) + (size_t)(2 * lane) * VPB + sp;
    o[0]   = a0;
    o[VPB] = a1;
}

// ---------------- kernel: zero the per-voxel counters ----------------
__global__ __launch_bounds__(256) void lss_zero_counts(uint32_t* __restrict__ counts) {
    int i = blockIdx.x * 256 + threadIdx.x;
    if (i < NVOX) counts[i] = 0u;
}

// ---------------- kernel: voxelize points, count + bin ----------------
__global__ __launch_bounds__(256) void lss_count_bin(const float* __restrict__ geom,
                                                     uint32_t* __restrict__ counts,
                                                     uint32_t* __restrict__ pflat,
                                                     uint32_t* __restrict__ bins) {
    int p = blockIdx.x * 256 + threadIdx.x;
    if (p >= NPTS) return;
    float px = geom[(size_t)p * 3 + 0];
    float py = geom[(size_t)p * 3 + 1];
    float pz = geom[(size_t)p * 3 + 2];
    // reference: ((pt - (bx - dx/2)) / dx).astype(int32)  -> trunc toward zero
    int ix = (int)((px + 50.0f) * 2.0f);      // /0.5 == *2 (exact, pow2)
    int iy = (int)((py + 50.0f) * 2.0f);
    int iz = (int)((pz + 10.0f) / 20.0f);
    bool kept = (ix >= 0) & (ix < NXY) & (iy >= 0) & (iy < NXY) & (iz == 0);

    uint32_t rec = DROPPED;
    if (kept) {
        uint32_t flat = (uint32_t)(p / PPB) * VPB + (uint32_t)ix * NXY + (uint32_t)iy;
        uint32_t slot = atomicAdd(&counts[flat], 1u);
        if (slot < CAP) {
            bins[(size_t)flat * CAP + slot] = (uint32_t)p;
            rec = flat;
        } else {
            rec = flat | OVF;   // exact fallback handled by lss_overflow_fix
        }
    }
    pflat[p] = rec;
}

// ---------------- kernel: exact fallback for bin-overflow points ----------------
__global__ __launch_bounds__(256) void lss_overflow_fix(const float* __restrict__ x,
                                                        const uint32_t* __restrict__ pflat,
                                                        float* __restrict__ out) {
    int p = blockIdx.x * 256 + threadIdx.x;
    if (p >= NPTS) return;
    uint32_t rec = pflat[p];
    if (!(rec & OVF)) return;                 // DROPPED has OVF clear
    uint32_t flat = rec & 0x7FFFFFFFu;
    uint32_t b  = flat / VPB;
    uint32_t sp = flat % VPB;
    const float* row = x + (size_t)p * NC;
    float* o = out + (size_t)b * ((size_t)NC * VPB) + sp;
#pragma unroll 4
    for (int c = 0; c < NC; ++c)
        atomicAdd(o + (size_t)c * VPB, row[c]);
}

// ---------------- launcher ----------------
extern "C" void kernel_launch(void* const* d_in, const int* in_sizes, int n_in,
                              void* d_out, int out_size, void* d_ws, size_t ws_size,
                              hipStream_t stream) {
    const float* geom = (const float*)d_in[0];   // geom_feats [B,N,D,H,W,3] f32
    const float* x    = (const float*)d_in[1];   // x          [B,N,D,H,W,C] f32
    float* out = (float*)d_out;                  // [4, 64, 200, 200] f32

    // workspace layout (all 16B aligned): counts | pflat | bins  (~23.9 MB)
    const size_t counts_bytes = (size_t)NVOX * 4;            // 640,000
    const size_t pflat_bytes  = (size_t)NPTS * 4;            // 2,770,944
    uint32_t* counts = (uint32_t*)d_ws;
    uint32_t* pflat  = (uint32_t*)((char*)d_ws + counts_bytes);
    uint32_t* bins   = (uint32_t*)((char*)d_ws + counts_bytes + pflat_bytes);

    lss_zero_counts  <<<(NVOX + 255) / 256, 256, 0, stream>>>(counts);
    lss_count_bin    <<<NPTS / 256,         256, 0, stream>>>(geom, counts, pflat, bins);
    lss_gather_reduce<<<NVOX / 8,           256, 0, stream>>>(x, counts, bins, out);
    lss_overflow_fix <<<NPTS / 256,         256, 0, stream>>>(x, pflat, out);
}